// LocalMap_25786983645304
// MI455X (gfx1250) — compile-verified
//
#include <hip/hip_runtime.h>

typedef __attribute__((ext_vector_type(2))) float v2f;
typedef __attribute__((ext_vector_type(4))) float v4f;
typedef __attribute__((ext_vector_type(8))) float v8f;

#define N_EXPERTS 8
#define D 512               // D_IN == D_OUT == 512
#define LDS_STRIDE 516      // 512 + 4 floats pad: conflict-free A-frag reads
#define MTILE 64            // tokens per block (4 WMMA M-subtiles)

// ---------------- phase 0: zero expert counters ----------------
__global__ void moe_init_kernel(int* ws) {
    if (threadIdx.x < N_EXPERTS) ws[threadIdx.x] = 0;   // counts
}

// ---------------- phase 1: histogram + copy z to output ----------------
__global__ void moe_count_kernel(const int* __restrict__ z, int* __restrict__ ws,
                                 int* __restrict__ outz, int T) {
    int t = blockIdx.x * blockDim.x + threadIdx.x;
    if (t < T) {
        int e = z[t];
        outz[t] = e;
        atomicAdd(&ws[e], 1);
    }
}

// ---------------- phase 2: 8-entry exclusive scan ----------------
// ws[0..7]=counts, ws[8..15]=cursors, ws[16..23]=starts, ws[32..]=tokenIds
__global__ void moe_scan_kernel(int* ws) {
    if (threadIdx.x == 0) {
        int acc = 0;
        for (int e = 0; e < N_EXPERTS; ++e) {
            int c = ws[e];
            ws[16 + e] = acc;   // start
            ws[8 + e]  = acc;   // cursor
            acc += c;
        }
    }
}

// ---------------- phase 3: scatter token ids into per-expert lists ----------------
__global__ void moe_scatter_kernel(const int* __restrict__ z, int* __restrict__ ws, int T) {
    int t = blockIdx.x * blockDim.x + threadIdx.x;
    if (t < T) {
        int pos = atomicAdd(&ws[8 + z[t]], 1);
        ws[32 + pos] = t;
    }
}

// ---------------- phase 4: grouped GEMM via V_WMMA_F32_16X16X4_F32 ----------------
// Block = 256 threads (8 wave32): 64 tokens (one expert) x 128 out cols.
// Each wave: one 16-wide N slice, 4 M-subtiles sharing each B fragment
// (4x B reuse -> 2 global loads per 4 WMMAs).
__global__ __launch_bounds__(256) void moe_gemm_kernel(
    const float* __restrict__ x, const float* __restrict__ W,
    const float* __restrict__ bias, const int* __restrict__ ws,
    float* __restrict__ y) {

    __shared__ __align__(16) float ax[MTILE * LDS_STRIDE];   // 132 KB of 320 KB WGP LDS
    __shared__ int toks[MTILE];

    const int* counts   = ws;
    const int* starts   = ws + 16;
    const int* tokenIds = ws + 32;

    // Map flat blockIdx.x -> (expert, M-tile within expert). Uniform per block.
    int bid = blockIdx.x;
    int expert = -1, mtile = 0, acc = 0;
    for (int e = 0; e < N_EXPERTS; ++e) {
        int nt = (counts[e] + (MTILE - 1)) >> 6;
        if (bid < acc + nt) { expert = e; mtile = bid - acc; break; }
        acc += nt;
    }
    if (expert < 0) return;   // uniform exit, before any barrier

    const int tid     = threadIdx.x;
    const int cnt     = counts[expert];
    const int start   = starts[expert];
    const int rowbase = mtile * MTILE;
    const int valid   = cnt - rowbase;        // uniform; >= MTILE except last tile

    if (tid < MTILE) {
        int r = rowbase + tid;
        toks[tid] = (r < cnt) ? tokenIds[start + r] : -1;
    }
    __syncthreads();

    // Gather 64x512 fp32 A tile into LDS (4 threads/row, 32 x 16B each).
    // x is streamed once -> non-temporal loads keep W hot in L2.
    {
        int row = tid >> 2;          // 0..63
        int sub = tid & 3;           // 0..3 -> contiguous 128-float chunk
        int tok = toks[row];
        v4f* dst = (v4f*)&ax[row * LDS_STRIDE + sub * 128];
        if (tok >= 0) {
            const v4f* src = (const v4f*)(x + (size_t)tok * D + sub * 128);
            #pragma unroll
            for (int j = 0; j < 32; ++j)
                dst[j] = __builtin_nontemporal_load(&src[j]);
        } else {
            v4f zv = {0.f, 0.f, 0.f, 0.f};
            #pragma unroll
            for (int j = 0; j < 32; ++j) dst[j] = zv;   // pad rows -> 0
        }
    }
    __syncthreads();

    const int lane = tid & 31;       // wave32
    const int wave = tid >> 5;       // 0..7 -> 16-wide N slice
    const int half = lane >> 4;      // 0/1  (K pair select)
    const int lm   = lane & 15;      // row (A) / col (B)

    const int nbase = blockIdx.y * 128 + wave * 16;

    // B fragment source: W[expert][k + 2*half + {0,1}][nbase + lm]  (L2-resident)
    const float* Wp = W + (size_t)expert * D * D + (size_t)(2 * half) * D + nbase + lm;

    // A fragment base per M-subtile (LDS): ax[s*16 + lm][k + 2*half + {0,1}]
    const float* ap[4];
    #pragma unroll
    for (int s = 0; s < 4; ++s)
        ap[s] = &ax[(s * 16 + lm) * LDS_STRIDE + 2 * half];

    v8f c[4] = {v8f{}, v8f{}, v8f{}, v8f{}};

    #pragma unroll 2
    for (int k = 0; k < D; k += 4) {
        v2f b;
        b.x = Wp[(size_t)k * D];
        b.y = Wp[(size_t)k * D + D];
        #pragma unroll
        for (int s = 0; s < 4; ++s) {
            v2f a = *(const v2f*)(ap[s] + k);      // ds_load_b64, bank-spread
            // D = A(16x4) * B(4x16) + C ; fp32 in/out, RNE
            c[s] = __builtin_amdgcn_wmma_f32_16x16x4_f32(
                       false, a, false, b, (short)0, c[s], false, false);
        }
    }

    // Bias: N = lm for every C VGPR -> one scalar per lane.
    const float bv = bias[expert * D + nbase + lm];

    // C layout: VGPR j holds rows M=j (lanes 0-15) and M=j+8 (lanes 16-31).
    // y is written once -> non-temporal stores.
    if (valid >= MTILE) {
        // Fast path (all but the last tile of each expert): no per-row masking.
        #pragma unroll
        for (int s = 0; s < 4; ++s) {
            #pragma unroll
            for (int j = 0; j < 8; ++j) {
                int tok = toks[s * 16 + j + 8 * half];
                __builtin_nontemporal_store(c[s][j] + bv,
                                            &y[(size_t)tok * D + nbase + lm]);
            }
        }
    } else {
        #pragma unroll
        for (int s = 0; s < 4; ++s) {
            #pragma unroll
            for (int j = 0; j < 8; ++j) {
                int tok = toks[s * 16 + j + 8 * half];
                if (tok >= 0)
                    __builtin_nontemporal_store(c[s][j] + bv,
                                                &y[(size_t)tok * D + nbase + lm]);
            }
        }
    }
}

extern "C" void kernel_launch(void* const* d_in, const int* in_sizes, int n_in,
                              void* d_out, int out_size, void* d_ws, size_t ws_size,
                              hipStream_t stream) {
    const int*   z = (const int*)d_in[0];
    const float* x = (const float*)d_in[1];
    const float* W = (const float*)d_in[2];
    const float* b = (const float*)d_in[3];
    const int T = in_sizes[0];          // B*N tokens

    int*   ws   = (int*)d_ws;           // [0..7] counts, [8..15] cursors,
                                        // [16..23] starts, [32..32+T) tokenIds
    int*   outz = (int*)d_out;          // z copied bit-exact
    float* y    = (float*)d_out + T;    // y follows z in the flat tuple

    moe_init_kernel<<<1, 32, 0, stream>>>(ws);
    moe_count_kernel<<<(T + 255) / 256, 256, 0, stream>>>(z, ws, outz, T);
    moe_scan_kernel<<<1, 1, 0, stream>>>(ws);
    moe_scatter_kernel<<<(T + 255) / 256, 256, 0, stream>>>(z, ws, T);

    const int maxTiles = T / MTILE + N_EXPERTS;   // ceil-tile worst case
    dim3 grid(maxTiles, D / 128);
    moe_gemm_kernel<<<grid, 256, 0, stream>>>(x, W, b, ws, y);
}